// YoloLoss_33938831573233
// MI455X (gfx1250) — compile-verified
//
#include <hip/hip_runtime.h>
#include <hip/hip_bf16.h>

// Problem constants (match reference)
#define BB 256
#define WW 19
#define HH 19
#define AA 5
#define CC 80
#define TT 20
#define NC (BB*WW*HH*AA)      // 462080 cell-anchor slots
#define NTOT (NC*CC)          // 36,966,400 cls_score elements

static constexpr int THREADS      = 256;
static constexpr int SMALL_BLOCKS = 256;
static constexpr int SCORE_BLOCKS = 512;

typedef __attribute__((ext_vector_type(2))) float v2f;
typedef __attribute__((ext_vector_type(8))) float v8f;

__device__ __forceinline__ float sigf(float x) { return 1.0f / (1.0f + expf(-x)); }

// ---------------- init workspace (mask/iou/tbbox/tidx + partials) ----------------
__global__ void k_init(float4* tb, float* obj, float* iou, int* tidx,
                       float* p_small, float* p_score) {
    int g = blockIdx.x * blockDim.x + threadIdx.x;
    int stride = gridDim.x * blockDim.x;
    for (int i = g; i < NC; i += stride) {
        tb[i]   = make_float4(0.f, 0.f, 0.f, 0.f);
        obj[i]  = 0.f;
        iou[i]  = 0.f;
        tidx[i] = -1;
    }
    if (g < SMALL_BLOCKS) p_small[g] = 0.f;
    if (g < SCORE_BLOCKS) p_score[g] = 0.f;
}

// ---------------- target assignment scatter (B*T*A threads, collision-free) ------
__global__ void k_scatter(const float* __restrict__ pred, const float* __restrict__ tobj,
                          const float* __restrict__ fmlim,
                          float4* tb, float* obj, float* iou_ws, int* tidx) {
    int idx = blockIdx.x * blockDim.x + threadIdx.x;
    if (idx >= BB * TT * AA) return;
    int a = idx % AA; int bt = idx / AA; int t = bt % TT; int b = bt / TT;
    const float inv = 1.0f / 32.0f;                       // 1/DS
    const float* to = tobj + (b * TT + t) * 4;
    float tox = to[0] * inv, toy = to[1] * inv, tw = to[2] * inv, th = to[3] * inv;
    int ii = (int)floorf(tox), jj = (int)floorf(toy);     // in [0,W),[0,H) by construction
    int cell = ((b * WW + ii) * HH + jj) * AA + a;
    const float* p = pred + cell * 5;
    int fm = (ii * HH + jj) * 2;
    float pw = sigf(p[2]) * fmlim[fm + 0];
    float ph = sigf(p[3]) * fmlim[fm + 1];
    float inter = fminf(pw, tw) * fminf(ph, th);          // center-aligned IoU
    float uni   = pw * ph + tw * th - inter;
    float iou   = inter / uni;
    if (iou > 0.5f) {                                     // IOU_THR
        obj[cell]    = 1.0f;
        iou_ws[cell] = iou;
        tb[cell]     = make_float4(tox, toy, tw, th);
        tidx[cell]   = t;
    }
}

// ---------------- noobj + obj + prior + true losses over pred_object -------------
__global__ void k_small(const float* __restrict__ pred, const float* __restrict__ anchors,
                        const float* __restrict__ fmc, const float* __restrict__ fmlim,
                        const int* __restrict__ epoch_p,
                        const float4* __restrict__ tb, const float* __restrict__ obj,
                        const float* __restrict__ iou_ws, float* p_small) {
    __shared__ float sdata[THREADS];
    int g = blockIdx.x * blockDim.x + threadIdx.x;
    int stride = gridDim.x * blockDim.x;
    float need_prior = (epoch_p[0] < 10) ? 1.0f : 0.0f;   // ANCHOR_TRAIN_EPOCHS
    const float KN = 0.25f  / (float)NC;                  // SCALE_NOOBJ*0.5 / NC
    const float KO = 2.5f   / (float)NC;                  // SCALE_OBJ*0.5   / NC
    const float KP = (1.25f / (float)NC) * need_prior;    // SCALE_OBJ*0.5   / (NC*2)
    const float KT = 0.625f / (float)NC;                  // SCALE_OBJ*0.5   / (NC*4)
    float acc = 0.f;
    for (int cell = g; cell < NC; cell += stride) {
        int a = cell % AA; int r = cell / AA; int j = r % HH; r /= HH; int i = r % WW;
        const float* p = pred + cell * 5;
        float px = sigf(p[0]), py = sigf(p[1]);
        float pw = sigf(p[2]), ph = sigf(p[3]);
        float conf = sigf(p[4]);
        int fm = (i * HH + j) * 2;
        px += fmc[fm + 0]; py += fmc[fm + 1];
        pw *= fmlim[fm + 0]; ph *= fmlim[fm + 1];
        float m  = obj[cell];
        float iv = iou_ws[cell];
        float4 t = tb[cell];
        float ax = anchors[a * 2 + 0], ay = anchors[a * 2 + 1];
        acc += KN * (1.f - m) * conf * conf;
        float dc = conf - iv;
        acc += KO * m * dc * dc;
        float dw = pw - ax, dh = ph - ay;
        acc += KP * m * (dw * dw + dh * dh);
        float d0 = px - t.x, d1 = py - t.y, d2 = pw - t.z, d3 = ph - t.w;
        acc += KT * m * (d0 * d0 + d1 * d1 + d2 * d2 + d3 * d3);
    }
    sdata[threadIdx.x] = acc;
    __syncthreads();
    for (int s = THREADS / 2; s > 0; s >>= 1) {
        if (threadIdx.x < s) sdata[threadIdx.x] += sdata[threadIdx.x + s];
        __syncthreads();
    }
    if (threadIdx.x == 0) p_small[blockIdx.x] = sdata[0];
}

// ---------------- score loss over cls_score (b128 stream + WMMA accumulation) ----
// C=80 is a multiple of 4, so every aligned float4 of cls_score lies in ONE cell:
// one tidx lookup + one /80 per 4 elements. Target row reconstructed branchlessly.
__global__ void k_score(const float4* __restrict__ cls4, const float* __restrict__ tl,
                        const int* __restrict__ tidx, float* p_score) {
    __shared__ float sw[THREADS / 32];
    const int N4 = NTOT / 4;                              // 9,241,600 float4 groups
    int g = blockIdx.x * blockDim.x + threadIdx.x;
    int stride = gridDim.x * blockDim.x;                  // 131072 lanes
    int iters = (N4 + stride - 1) / stride;               // UNIFORM -> EXEC all-1s at WMMA
    v8f acc = {0.f, 0.f, 0.f, 0.f, 0.f, 0.f, 0.f, 0.f};
    v2f ones; ones.x = 1.f; ones.y = 1.f;                 // B = all-ones 4x16 matrix
    for (int it = 0; it < iters; ++it) {
        int q  = g + it * stride;
        int qc = min(q, N4 - 1);                          // clamp tail: loads stay valid
        int e0   = qc * 4;
        int cell = e0 / CC;                               // same cell for all 4 elems
        int c0   = e0 - cell * CC;                        // multiple of 4
        float4 cv = cls4[qc];                             // global_load_b128, coalesced
        __builtin_prefetch((const float*)(cls4 + qc + stride), 0, 0); // global_prefetch_b8
        int t   = tidx[cell];
        bool hit = (t >= 0);
        int b   = cell / (WW * HH * AA);
        int tcl = hit ? t : 0;                            // clamped -> always-valid gather
        float4 tv = *(const float4*)(tl + (b * TT + tcl) * CC + c0);
        float dw  = (c0 == CC - 4) ? 1.0f : 0.0f;         // default one-hot at class C-1
        float t0 = hit ? tv.x : 0.f;
        float t1 = hit ? tv.y : 0.f;
        float t2 = hit ? tv.z : 0.f;
        float t3 = hit ? tv.w : dw;
        float r0 = cv.x - t0, r1 = cv.y - t1, r2 = cv.z - t2, r3 = cv.w - t3;
        float s0 = fmaf(r0, r0, r1 * r1);
        float s1 = fmaf(r2, r2, r3 * r3);
        bool valid = (q < N4);
        v2f a; a.x = valid ? s0 : 0.f; a.y = valid ? s1 : 0.f;  // A = 16x4 partials
        // D += A x ones : every A element counted 16x across the 16 output columns
        acc = __builtin_amdgcn_wmma_f32_16x16x4_f32(
            /*neg_a=*/false, a, /*neg_b=*/false, ones,
            /*c_mod=*/(short)0, acc, /*reuse_a=*/false, /*reuse_b=*/false);
    }
    float ssum = acc[0] + acc[1] + acc[2] + acc[3] + acc[4] + acc[5] + acc[6] + acc[7];
    for (int off = 16; off > 0; off >>= 1)
        ssum += __shfl_xor(ssum, off, 32);                // wave32 reduction
    int wid = threadIdx.x >> 5;
    if ((threadIdx.x & 31) == 0) sw[wid] = ssum;
    __syncthreads();
    if (threadIdx.x == 0) {
        float bs = 0.f;
        for (int w = 0; w < THREADS / 32; ++w) bs += sw[w];
        const float scale = (2.5f / (float)NTOT) * (1.0f / 16.0f); // SCALE_OBJ*0.5/mean, /16 dup
        p_score[blockIdx.x] = bs * scale;
    }
}

// ---------------- deterministic final reduction ----------------------------------
__global__ void k_final(const float* __restrict__ p_small, const float* __restrict__ p_score,
                        float* out) {
    if (blockIdx.x == 0 && threadIdx.x == 0) {
        float s = 0.f;
        for (int i = 0; i < SMALL_BLOCKS; ++i) s += p_small[i];
        for (int i = 0; i < SCORE_BLOCKS; ++i) s += p_score[i];
        out[0] = s * 0.25f;                               // (sum of 5 means) / 4
    }
}

extern "C" void kernel_launch(void* const* d_in, const int* in_sizes, int n_in,
                              void* d_out, int out_size, void* d_ws, size_t ws_size,
                              hipStream_t stream) {
    const int*   epoch   = (const int*)d_in[0];
    const float* cls     = (const float*)d_in[1];
    const float* pred    = (const float*)d_in[2];
    const float* tobj    = (const float*)d_in[3];
    const float* tl      = (const float*)d_in[4];
    const float* anchors = (const float*)d_in[5];
    const float* fmc     = (const float*)d_in[6];
    const float* fmlim   = (const float*)d_in[7];

    char* ws = (char*)d_ws;
    float4* tb     = (float4*)ws;              size_t off = (size_t)NC * 16;  // 16B-aligned first
    float*  obj    = (float*)(ws + off);       off += (size_t)NC * 4;
    float*  iou    = (float*)(ws + off);       off += (size_t)NC * 4;
    int*    tidx   = (int*)(ws + off);         off += (size_t)NC * 4;
    float*  p_small = (float*)(ws + off);      off += (size_t)SMALL_BLOCKS * 4;
    float*  p_score = (float*)(ws + off);
    float*  out = (float*)d_out;

    k_init<<<(NC + THREADS - 1) / THREADS, THREADS, 0, stream>>>(tb, obj, iou, tidx, p_small, p_score);
    k_scatter<<<(BB * TT * AA + THREADS - 1) / THREADS, THREADS, 0, stream>>>(
        pred, tobj, fmlim, tb, obj, iou, tidx);
    k_small<<<SMALL_BLOCKS, THREADS, 0, stream>>>(
        pred, anchors, fmc, fmlim, epoch, tb, obj, iou, p_small);
    k_score<<<SCORE_BLOCKS, THREADS, 0, stream>>>((const float4*)cls, tl, tidx, p_score);
    k_final<<<1, 32, 0, stream>>>(p_small, p_score, out);
}